// LocallyConnected3D_17128329576944
// MI455X (gfx1250) — compile-verified
//
#include <hip/hip_runtime.h>
#include <hip/hip_bf16.h>

// LocallyConnected3D: x[16,24,24,24,16] (f32), kernel[10648,432,32] (f32),
// bias[22,22,22,32] (f32) -> out[16,22,22,22,32] (f32).
// One wave32 per output location p. Per-p GEMM: [16x432] x [432x32] via
// V_WMMA_F32_16X16X4_F32, K reduced in 108 chunks of 4, N=32 as two 16-wide
// accumulator tiles. Memory-bound on the 589 MB one-shot weight stream ->
// weight loads are non-temporal; x (14 MB) stays hot in the 192 MB L2.

typedef __attribute__((ext_vector_type(2))) float v2f;
typedef __attribute__((ext_vector_type(8))) float v8f;

#define NP   10648      // 22*22*22 output locations
#define KF   432        // 27 * 16 patch features
#define NF   32         // output channels
#define NB   16         // batch
#define DIN  24
#define CIN  16

__global__ __launch_bounds__(256) void lc3d_wmma_f32(
    const float* __restrict__ x,
    const float* __restrict__ wgt,
    const float* __restrict__ bias,
    float* __restrict__ out)
{
    const int wave = threadIdx.x >> 5;
    const int lane = threadIdx.x & 31;
    const int p    = blockIdx.x * 8 + wave;   // grid sized so p < NP always

    // decompose p -> (od, oh, ow)
    const int od  = p / (22 * 22);
    const int rem = p - od * (22 * 22);
    const int oh  = rem / 22;
    const int ow  = rem - oh * 22;

    const int t = lane >> 4;     // lane half: selects K sub-pair {0,1} vs {2,3}
    const int n = lane & 15;     // A: batch row M ; B: output column N

    const float* __restrict__ wp = wgt + (size_t)p * (KF * NF);

    v8f acc0 = {};   // columns f = 0..15
    v8f acc1 = {};   // columns f = 16..31

    for (int s = 0; s < 27; ++s) {                 // kernel spatial offsets
        const int di = s / 9;
        const int dj = (s / 3) % 3;
        const int dk = s % 3;

        // A source: x[b = n, od+di, oh+dj, ow+dk, c]; channels contiguous.
        const float* __restrict__ xb =
            x + ((((size_t)n * DIN + (od + di)) * DIN + (oh + dj)) * DIN
                 + (ow + dk)) * CIN + t * 2;

        // B source rows for this lane: kr = s*16 + cb + t*2, column n.
        const float* __restrict__ wrow = wp + ((size_t)s * 16 + t * 2) * NF + n;

        #pragma unroll
        for (int cb = 0; cb < 16; cb += 4) {       // 4 k-chunks per spatial tap
            // A fragment: 2 contiguous f32 per lane (8B aligned)
            const v2f a = *(const v2f*)(xb + cb);

            // B fragments, non-temporal (weights are single-use)
            v2f b0, b1;
            b0.x = __builtin_nontemporal_load(wrow + cb * NF);
            b0.y = __builtin_nontemporal_load(wrow + cb * NF + NF);
            b1.x = __builtin_nontemporal_load(wrow + cb * NF + 16);
            b1.y = __builtin_nontemporal_load(wrow + cb * NF + NF + 16);

            acc0 = __builtin_amdgcn_wmma_f32_16x16x4_f32(
                       false, a, false, b0, (short)0, acc0, false, false);
            acc1 = __builtin_amdgcn_wmma_f32_16x16x4_f32(
                       false, a, false, b1, (short)0, acc1, false, false);
        }
    }

    // bias depends only on (p, f); broadcast across the M (batch) dimension
    const float bv0 = bias[p * NF + n];
    const float bv1 = bias[p * NF + 16 + n];

    // D layout: VGPR r, lane L -> M = r + 8*t, N = n
    #pragma unroll
    for (int r = 0; r < 8; ++r) {
        const int m = r + 8 * t;
        float* __restrict__ o = out + ((size_t)m * NP + p) * NF + n;
        o[0]  = acc0[r] + bv0;
        o[16] = acc1[r] + bv1;
    }
}

extern "C" void kernel_launch(void* const* d_in, const int* in_sizes, int n_in,
                              void* d_out, int out_size, void* d_ws, size_t ws_size,
                              hipStream_t stream)
{
    const float* x    = (const float*)d_in[0];  // [16,24,24,24,16]
    const float* wgt  = (const float*)d_in[1];  // [10648,432,32]
    const float* bias = (const float*)d_in[2];  // [22,22,22,32]
    float* out        = (float*)d_out;          // [16,22,22,22,32]

    // 8 waves per block, exactly 10648 waves total (1331 * 8)
    dim3 grid(NP / 8);
    dim3 block(256);
    lc3d_wmma_f32<<<grid, block, 0, stream>>>(x, wgt, bias, out);
}